// AdjointCircuitModule_31662498906415
// MI455X (gfx1250) — compile-verified
//
#include <hip/hip_runtime.h>
#include <hip/hip_bf16.h>

#define NQ      13
#define DIM     8192           // 2^13
#define NLAYERS 8

typedef float v2f __attribute__((ext_vector_type(2)));
typedef float v8f __attribute__((ext_vector_type(8)));

// ---------------------------------------------------------------------------
// Kernel A: full state-vector simulation in LDS. One workgroup, 1024 threads,
// 8 amplitudes per thread per pass.
//   - RX layer: 5 register-tile passes (3 local bits each; RX gates commute)
//   - RZ layer: fused into one diagonal phase application (all 13 qubits)
//   - CNOT chain: single permutation pass, x = y ^ (y >> 1)  (Gray code)
// ---------------------------------------------------------------------------
extern "C" __global__ __launch_bounds__(1024)
void sim_kernel(const float* __restrict__ params, float* __restrict__ out_rs)
{
    extern __shared__ float lds[];
    float* re = lds;          // DIM floats
    float* im = lds + DIM;    // DIM floats
    const int t = threadIdx.x;

    // init |0...0>
    for (int k = 0; k < 8; ++k) {
        const int i = t * 8 + k;
        re[i] = (i == 0) ? 1.0f : 0.0f;
        im[i] = 0.0f;
    }
    __syncthreads();

    float ar[8], ai[8];
    int   idx[8];

    for (int layer = 0; layer < NLAYERS; ++layer) {
        const int pbase = layer * 26;      // 13 RX params then 13 RZ params

        // ---- RX passes (groups of 3 local flat-bits) + fused RZ in last ----
        for (int g = 0; g < 5; ++g) {
            if (g < 4) {
                const int sh  = 3 * g;
                const int low = t & ((1 << sh) - 1);
                const int hi  = (t >> sh) << (sh + 3);
                for (int k = 0; k < 8; ++k) idx[k] = hi | (k << sh) | low;
            } else {
                // local bits {0,1,12}: bit12 is local bit 2
                for (int k = 0; k < 8; ++k)
                    idx[k] = ((k >> 2) << 12) | (t << 2) | (k & 3);
            }
            for (int k = 0; k < 8; ++k) { ar[k] = re[idx[k]]; ai[k] = im[idx[k]]; }

            if (g < 4) {
                // RX on qubits q = 12-(3g+j), j=0..2 (flat bit 3g+j)
                for (int j = 0; j < 3; ++j) {
                    const int   q = 12 - (3 * g + j);
                    const float h = 0.5f * params[pbase + q];
                    const float c = __cosf(h), s = __sinf(h);
                    const int   jm = 1 << j;
                    for (int k = 0; k < 8; ++k) {
                        if (k & jm) continue;
                        const int k1 = k | jm;
                        const float arA = ar[k],  aiA = ai[k];
                        const float arB = ar[k1], aiB = ai[k1];
                        // RX: a' = c*a - i s*b ; b' = -i s*a + c*b
                        ar[k]  = c * arA + s * aiB;
                        ai[k]  = c * aiA - s * arB;
                        ar[k1] = c * arB + s * aiA;
                        ai[k1] = c * aiB - s * arA;
                    }
                }
            } else {
                // RX on qubit 0 (local bit 2)
                {
                    const float h = 0.5f * params[pbase + 0];
                    const float c = __cosf(h), s = __sinf(h);
                    for (int k = 0; k < 4; ++k) {
                        const int k1 = k | 4;
                        const float arA = ar[k],  aiA = ai[k];
                        const float arB = ar[k1], aiB = ai[k1];
                        ar[k]  = c * arA + s * aiB;
                        ai[k]  = c * aiA - s * arB;
                        ar[k1] = c * arB + s * aiA;
                        ai[k1] = c * aiB - s * arA;
                    }
                }
                // fused RZ layer: diag phase e^{i*ang}, ang = sum_q (+-)h_q
                float hz[13];
                for (int q = 0; q < 13; ++q) hz[q] = 0.5f * params[pbase + 13 + q];
                for (int k = 0; k < 8; ++k) {
                    const int id = idx[k];
                    float ang = 0.0f;
                    for (int q = 0; q < 13; ++q)
                        ang += ((id >> (12 - q)) & 1) ? hz[q] : -hz[q];
                    const float c = __cosf(ang), s = __sinf(ang);
                    const float r0 = ar[k], i0 = ai[k];
                    ar[k] = r0 * c - i0 * s;
                    ai[k] = i0 * c + r0 * s;
                }
            }
            // in-place safe: each thread writes exactly the indices it read
            for (int k = 0; k < 8; ++k) { re[idx[k]] = ar[k]; im[idx[k]] = ai[k]; }
            __syncthreads();
        }

        // ---- CNOT chain: state'[y] = state[y ^ (y>>1)] ----
        for (int k = 0; k < 8; ++k) {
            const int y = t * 8 + k;
            const int x = y ^ (y >> 1);
            ar[k] = re[x]; ai[k] = im[x];
        }
        __syncthreads();   // all gathers complete before any overwrite
        for (int k = 0; k < 8; ++k) {
            const int y = t * 8 + k;
            re[y] = ar[k]; im[y] = ai[k];
        }
        __syncthreads();
    }

    // export r (= Re psi) and s (= Im psi)
    for (int k = 0; k < 8; ++k) {
        const int i = t * 8 + k;
        out_rs[i]       = re[i];
        out_rs[DIM + i] = im[i];
    }
}

// ---------------------------------------------------------------------------
// Kernel B: loss partials = r^T O r + s^T O s via V_WMMA_F32_16X16X4_F32.
// P = O * W, W = [r s 0...0] (8192x16). One wave per (16-row tile, K-quarter).
// A lane layout (ISA 7.12.2, 32-bit 16x4 A): lane&15 = row M, lane>>4 picks
// K-pair -> one b64 global load per lane per WMMA. B's zero columns come from
// a zero-filled LDS region selected per-lane ONCE. A is software-pipelined
// through two 16-load register buffers so ~16 b64 loads (4 KB/wave) stay in
// flight -> enough MLP to cover HBM/L2 latency at 6 waves/SIMD.
// ---------------------------------------------------------------------------
#define WAVES_PER_BLOCK 8
#define KSPLIT 4
#define NTILES (DIM / 16)                       // 512
#define NWAVES (NTILES * KSPLIT)                // 2048
#define NBLOCKS (NWAVES / WAVES_PER_BLOCK)      // 256

extern "C" __global__ __launch_bounds__(256)
void quadform_kernel(const float* __restrict__ O,
                     const float* __restrict__ rs,
                     float* __restrict__ partials)
{
    extern __shared__ float lds[];
    float* rL = lds;              // DIM floats: Re psi
    float* sL = lds + DIM;        // DIM floats: Im psi
    float* zL = lds + 2 * DIM;    // DIM floats: zeros (B columns 2..15)

    // cooperative load of psi (r,s) into LDS + zero the pad region
    {
        const float4* src = (const float4*)rs;
        float4*       dst = (float4*)lds;
        const float4  z4  = make_float4(0.f, 0.f, 0.f, 0.f);
        for (int i = threadIdx.x; i < (2 * DIM) / 4; i += blockDim.x)
            dst[i] = src[i];
        for (int i = threadIdx.x; i < DIM / 4; i += blockDim.x)
            dst[(2 * DIM) / 4 + i] = z4;
    }
    __syncthreads();

    const int wave  = threadIdx.x >> 5;
    const int lane  = threadIdx.x & 31;
    const int gwave = blockIdx.x * WAVES_PER_BLOCK + wave;   // 0..NWAVES-1
    const int tile  = gwave / KSPLIT;                        // 0..511
    const int quar  = gwave % KSPLIT;
    const int m     = tile * 16;
    const int n     = lane & 15;        // WMMA column / row-in-tile
    const int hi    = lane >> 4;        // K-pair selector
    const int k0    = quar * (DIM / KSPLIT);
    const int k1    = k0 + (DIM / KSPLIT);

    const float* Arow = O + (size_t)(m + n) * DIM + 2 * hi;  // A: row m+n
    // B: col0 = r, col1 = s, cols 2..15 read the zero region (loop-invariant)
    const float* bsrc = (n == 0) ? rL : ((n == 1) ? sL : zL);

    v8f c0 = {}, c1 = {}, c2 = {}, c3 = {};
    v2f aA[16], aB[16];    // double-buffered A stage: 64 floats (16 k-chunks) each

    auto load16 = [&](v2f* buf, int kk) {
        #pragma unroll
        for (int i = 0; i < 16; ++i)
            buf[i] = *(const v2f*)(Arow + kk + 4 * i);
    };
    auto wmma16 = [&](const v2f* buf, int kk) {
        #pragma unroll
        for (int g = 0; g < 4; ++g) {
            const int kc = kk + 16 * g + 2 * hi;
            v2f b0 = *(const v2f*)(bsrc + kc);
            v2f b1 = *(const v2f*)(bsrc + kc + 4);
            v2f b2 = *(const v2f*)(bsrc + kc + 8);
            v2f b3 = *(const v2f*)(bsrc + kc + 12);
            c0 = __builtin_amdgcn_wmma_f32_16x16x4_f32(false, buf[4*g+0], false, b0,
                                                       (short)0, c0, false, false);
            c1 = __builtin_amdgcn_wmma_f32_16x16x4_f32(false, buf[4*g+1], false, b1,
                                                       (short)0, c1, false, false);
            c2 = __builtin_amdgcn_wmma_f32_16x16x4_f32(false, buf[4*g+2], false, b2,
                                                       (short)0, c2, false, false);
            c3 = __builtin_amdgcn_wmma_f32_16x16x4_f32(false, buf[4*g+3], false, b3,
                                                       (short)0, c3, false, false);
        }
    };

    // software pipeline: while WMMAs consume one buffer, the other's 16 loads
    // are in flight. Uniform guard keeps the tail in-bounds (EXEC untouched).
    load16(aA, k0);
    for (int k = k0; k < k1; k += 128) {
        load16(aB, k + 64);
        wmma16(aA, k);
        if (k + 128 < k1) load16(aA, k + 128);
        wmma16(aB, k + 64);
    }

    const v8f cc = c0 + c1 + c2 + c3;

    // D layout: VGPR j holds rows m+j (lanes 0-15) and m+8+j (lanes 16-31).
    // Column 0 = (O r)[row], column 1 = (O s)[row]; dot with r / s.
    float part = 0.0f;
    if (n < 2) {
        const float* w = bsrc;
        #pragma unroll
        for (int j = 0; j < 8; ++j)
            part += cc[j] * w[m + 8 * hi + j];
    }
    #pragma unroll
    for (int off = 16; off > 0; off >>= 1)
        part += __shfl_xor(part, off, 32);
    if (lane == 0) partials[gwave] = part;
}

// ---------------------------------------------------------------------------
// Kernel C: deterministic final reduction (no atomics, overwrites d_out).
// ---------------------------------------------------------------------------
extern "C" __global__ __launch_bounds__(32)
void reduce_kernel(const float* __restrict__ partials, float* __restrict__ out)
{
    const int lane = threadIdx.x;
    float sum = 0.0f;
    for (int i = lane; i < NWAVES; i += 32) sum += partials[i];
    #pragma unroll
    for (int off = 16; off > 0; off >>= 1)
        sum += __shfl_xor(sum, off, 32);
    if (lane == 0) out[0] = sum;
}

// ---------------------------------------------------------------------------
extern "C" void kernel_launch(void* const* d_in, const int* in_sizes, int n_in,
                              void* d_out, int out_size, void* d_ws, size_t ws_size,
                              hipStream_t stream)
{
    const float* params = (const float*)d_in[0];   // [208] f32
    const float* O      = (const float*)d_in[1];   // [8192*8192] f32, row-major

    float* ws       = (float*)d_ws;
    float* rs       = ws;               // 2*DIM floats: r then s
    float* partials = ws + 2 * DIM;     // NWAVES floats

    sim_kernel<<<1, 1024, 2 * DIM * sizeof(float), stream>>>(params, rs);
    quadform_kernel<<<NBLOCKS, 256, 3 * DIM * sizeof(float), stream>>>(O, rs, partials);
    reduce_kernel<<<1, 32, 0, stream>>>(partials, (float*)d_out);
}